// StellarGNN_9938554323662
// MI455X (gfx1250) — compile-verified
//
#include <hip/hip_runtime.h>

#define HID 64
#define NGRAPH 64
#define OUTD 5

typedef __attribute__((ext_vector_type(2))) float v2f;
typedef __attribute__((ext_vector_type(8))) float v8f;

__device__ __forceinline__ void gaddf(float* p, float v) {
  __hip_atomic_fetch_add(p, v, __ATOMIC_RELAXED, __HIP_MEMORY_SCOPE_AGENT);
}
__device__ __forceinline__ void laddf(float* p, float v) {
  __hip_atomic_fetch_add(p, v, __ATOMIC_RELAXED, __HIP_MEMORY_SCOPE_WORKGROUP);
}

// deg[n] = 1 (self loop), zero graph sums/counts
__global__ void gcn_init(float* deg, float* gsum, float* gcnt, int n) {
  int i = blockIdx.x * blockDim.x + threadIdx.x;
  if (i < n) deg[i] = 1.0f;
  if (i < NGRAPH * HID) gsum[i] = 0.0f;
  if (i < NGRAPH) gcnt[i] = 0.0f;
}

__global__ void gcn_deg(const int* __restrict__ dst, float* deg, int E) {
  int e = blockIdx.x * blockDim.x + threadIdx.x;
  if (e < E) gaddf(&deg[dst[e]], 1.0f);
}

__global__ void gcn_rsqrt(float* deg, int n) {
  int i = blockIdx.x * blockDim.x + threadIdx.x;
  if (i < n) deg[i] = rsqrtf(deg[i]);
}

// per-edge symmetric norm, shared by both layers
__global__ void gcn_norm(const int* __restrict__ src, const int* __restrict__ dst,
                         const float* __restrict__ dinv, float* __restrict__ wedge, int E) {
  int e = blockIdx.x * blockDim.x + threadIdx.x;
  if (e < E) wedge[e] = dinv[src[e]] * dinv[dst[e]];
}

// t0[n][j] = sum_k x[n][k]*W1[k][j]; t1 = dinv^2 * t0 (fused self-loop init)
__global__ void gcn_gemm1(const float* __restrict__ x, const float* __restrict__ W1,
                          const float* __restrict__ dinv, float* __restrict__ t0,
                          float* __restrict__ t1, int n) {
  int i = blockIdx.x * blockDim.x + threadIdx.x;
  if (i >= n * HID) return;
  int node = i >> 6, j = i & 63;
  float s = x[node * 3 + 0] * W1[0 * HID + j]
          + x[node * 3 + 1] * W1[1 * HID + j]
          + x[node * 3 + 2] * W1[2 * HID + j];
  float d = dinv[node];
  t0[i] = s;
  t1[i] = d * d * s;
}

// thread = (edge, 4-col quad): b128 gather + 4 f32 scatter-atomics
__global__ void gcn_edge(const int* __restrict__ src, const int* __restrict__ dst,
                         const float* __restrict__ wedge, const float* __restrict__ tin,
                         float* __restrict__ tout, long long E) {
  long long gid = (long long)blockIdx.x * blockDim.x + threadIdx.x;
  if (gid >= E * 16) return;
  long long e = gid >> 4;
  int q = (int)(gid & 15) * 4;
  int s = src[e], d = dst[e];
  float w = wedge[e];
  const float4 v = *(const float4*)(tin + (long long)s * HID + q);
  float* o = tout + (long long)d * HID + q;
  gaddf(o + 0, w * v.x);
  gaddf(o + 1, w * v.y);
  gaddf(o + 2, w * v.z);
  gaddf(o + 3, w * v.w);
}

// hout = relu(hin + b1) @ W2 via V_WMMA_F32_16X16X4_F32; hsc = dinv^2 * hout fused.
// block = 128 threads = 4 waves; each wave computes a 16x64 output tile.
__global__ void gcn_gemm2(const float* __restrict__ hin, const float* __restrict__ b1,
                          const float* __restrict__ W2, const float* __restrict__ dinv,
                          float* __restrict__ hout, float* __restrict__ hsc, int n) {
  int wave = threadIdx.x >> 5;
  int lane = threadIdx.x & 31;
  int half = lane >> 4;       // 0: lanes 0-15, 1: lanes 16-31
  int mr = lane & 15;
  int m0 = blockIdx.x * 64 + wave * 16;
  int row = m0 + mr;
  int rowc = row < n ? row : (n - 1);  // clamp; garbage rows never stored

  v8f acc[4];
#pragma unroll
  for (int t = 0; t < 4; ++t) acc[t] = (v8f){0.f, 0.f, 0.f, 0.f, 0.f, 0.f, 0.f, 0.f};

#pragma unroll
  for (int s = 0; s < 16; ++s) {
    int k = 4 * s + half * 2;  // A layout: lanes 0-15 hold K pair {0,1}; lanes 16-31 {2,3}
    float a0 = fmaxf(hin[(long long)rowc * HID + k] + b1[k], 0.0f);
    float a1 = fmaxf(hin[(long long)rowc * HID + k + 1] + b1[k + 1], 0.0f);
    v2f a = {a0, a1};
#pragma unroll
    for (int t = 0; t < 4; ++t) {
      int ncol = t * 16 + mr;  // B layout mirrors A with N across lanes
      v2f b = {W2[k * HID + ncol], W2[(k + 1) * HID + ncol]};
      acc[t] = __builtin_amdgcn_wmma_f32_16x16x4_f32(
          false, a, false, b, (short)0, acc[t], false, false);
    }
  }

  // C/D layout: VGPR r -> M = r (lanes 0-15) / r+8 (lanes 16-31), N = lane&15
#pragma unroll
  for (int r = 0; r < 8; ++r) {
    int orow = m0 + r + half * 8;
    if (orow < n) {
      float dd = dinv[orow];
      dd = dd * dd;
#pragma unroll
      for (int t = 0; t < 4; ++t) {
        float val = acc[t][r];
        long long idx = (long long)orow * HID + t * 16 + mr;
        hout[idx] = val;
        hsc[idx] = dd * val;
      }
    }
  }
}

// per-block LDS accumulation of graph sums/counts, h = relu(t + b2)
__global__ void gcn_pool(const float* __restrict__ h, const float* __restrict__ b2,
                         const int* __restrict__ batch, float* gsum, float* gcnt, int n) {
  __shared__ float ls[NGRAPH * HID];
  __shared__ float lc[NGRAPH];
  int tid = threadIdx.x;  // 256 threads
  for (int i = tid; i < NGRAPH * HID; i += 256) ls[i] = 0.0f;
  if (tid < NGRAPH) lc[tid] = 0.0f;
  __syncthreads();

  int j = tid & 63;
  int sub = tid >> 6;  // 4 nodes per iteration
  int base = blockIdx.x * 1024;
  for (int it = 0; it < 256; ++it) {
    int node = base + it * 4 + sub;
    if (node < n) {
      int g = batch[node];
      float v = fmaxf(h[(long long)node * HID + j] + b2[j], 0.0f);
      laddf(&ls[g * HID + j], v);
      if (j == 0) laddf(&lc[g], 1.0f);
    }
  }
  __syncthreads();

  for (int i = tid; i < NGRAPH * HID; i += 256)
    if (ls[i] != 0.0f) gaddf(&gsum[i], ls[i]);
  if (tid < NGRAPH && lc[tid] != 0.0f) gaddf(&gcnt[tid], lc[tid]);
}

// out[g][o] = (sum_j gsum[g][j] * Wfc[j][o]) / max(cnt,1) + bfc[o]
__global__ void gcn_final(const float* __restrict__ gsum, const float* __restrict__ gcnt,
                          const float* __restrict__ Wfc, const float* __restrict__ bfc,
                          float* __restrict__ out) {
  int tid = threadIdx.x;
  if (tid >= NGRAPH * OUTD) return;
  int g = tid / OUTD, o = tid % OUTD;
  float inv = 1.0f / fmaxf(gcnt[g], 1.0f);
  float s = 0.0f;
#pragma unroll
  for (int j = 0; j < HID; ++j) s += gsum[g * HID + j] * Wfc[j * OUTD + o];
  out[tid] = s * inv + bfc[o];
}

extern "C" void kernel_launch(void* const* d_in, const int* in_sizes, int n_in,
                              void* d_out, int out_size, void* d_ws, size_t ws_size,
                              hipStream_t stream) {
  const float* x   = (const float*)d_in[0];
  const int*   ei  = (const int*)d_in[1];
  const int*   bat = (const int*)d_in[2];
  const float* W1  = (const float*)d_in[3];
  const float* b1  = (const float*)d_in[4];
  const float* W2  = (const float*)d_in[5];
  const float* b2  = (const float*)d_in[6];
  const float* Wfc = (const float*)d_in[7];
  const float* bfc = (const float*)d_in[8];
  float* out = (float*)d_out;

  const int n = in_sizes[0] / 3;        // 100000
  const int E = in_sizes[1] / 2;        // 1600000
  const int* src = ei;
  const int* dst = ei + E;

  float* ws    = (float*)d_ws;
  float* dinv  = ws;                              // [n] (deg, then rsqrt in place)
  float* wedge = dinv + n;                        // [E] per-edge norm
  float* t0    = wedge + E;                       // [n*64]
  float* t1    = t0 + (size_t)n * HID;            // [n*64]
  float* gsum  = t1 + (size_t)n * HID;            // [64*64]
  float* gcnt  = gsum + NGRAPH * HID;             // [64]

  int initN = n > NGRAPH * HID ? n : NGRAPH * HID;
  gcn_init<<<(initN + 255) / 256, 256, 0, stream>>>(dinv, gsum, gcnt, n);
  gcn_deg<<<(E + 255) / 256, 256, 0, stream>>>(dst, dinv, E);
  gcn_rsqrt<<<(n + 255) / 256, 256, 0, stream>>>(dinv, n);
  gcn_norm<<<(E + 255) / 256, 256, 0, stream>>>(src, dst, dinv, wedge, E);

  long long ew = (long long)E * 16;

  // layer 1: t0 = x@W1, t1 = self-loop init, then edge scatter into t1
  gcn_gemm1<<<(n * HID + 255) / 256, 256, 0, stream>>>(x, W1, dinv, t0, t1, n);
  gcn_edge<<<(unsigned)((ew + 255) / 256), 256, 0, stream>>>(src, dst, wedge, t0, t1, E);

  // layer 2: t0 = relu(t1+b1)@W2 (WMMA), t1 = self-loop init, edge scatter into t1
  gcn_gemm2<<<(n + 63) / 64, 128, 0, stream>>>(t1, b1, W2, dinv, t0, t1, n);
  gcn_edge<<<(unsigned)((ew + 255) / 256), 256, 0, stream>>>(src, dst, wedge, t0, t1, E);

  // pool + classifier
  gcn_pool<<<(n + 1023) / 1024, 256, 0, stream>>>(t1, b2, bat, gsum, gcnt, n);
  gcn_final<<<1, NGRAPH * OUTD, 0, stream>>>(gsum, gcnt, Wfc, bfc, out);
}